// CNN2DAttWind_NL_12309376270825
// MI455X (gfx1250) — compile-verified
//
#include <hip/hip_runtime.h>
#include <hip/hip_bf16.h>
#include <math.h>

// ---------------- CDNA5 WMMA types ----------------
typedef __attribute__((ext_vector_type(2))) float v2f;
typedef __attribute__((ext_vector_type(8))) float v8f;
typedef __attribute__((ext_vector_type(2))) int   i2v;

// ---------------- problem constants ----------------
#define BATCH   8192
#define CIN     16
#define HIN     6
#define HOUT    5
#define HW      25          // 5*5
#define FM      64
#define DK      32
#define DV      32
#define NH      32          // dkh = dvh = 1
#define KCONV   64          // CIN * 2 * 2  (im2col K)
#define HID     128
#define OUTCH   5

// kernel-1 tiling
#define GIMG    4                  // images per workgroup
#define COLS    (GIMG * HW)        // 100 columns
#define NCOLP   112                // 7 n-tiles * 16
#define PST     66                 // padded LDS row stride (floats), 8B aligned

// ---------------- CDNA5 async global->LDS path (guarded) ----------------
#if defined(__has_builtin)
#  if __has_builtin(__builtin_amdgcn_global_load_async_to_lds_b64) && \
      __has_builtin(__builtin_amdgcn_s_wait_asynccnt)
#    define ASYNC_B64 1
#  endif
#  if __has_builtin(__builtin_amdgcn_global_load_async_to_lds_b32) && \
      __has_builtin(__builtin_amdgcn_s_wait_asynccnt)
#    define ASYNC_B32 1
#  endif
#endif

// copy 8 bytes global -> LDS (async DMA when available, no VGPR bounce)
__device__ __forceinline__ void lds_copy_b64(const float* g, float* l)
{
#if defined(ASYNC_B64)
    __builtin_amdgcn_global_load_async_to_lds_b64(
        (__attribute__((address_space(1))) i2v*)(g),
        (__attribute__((address_space(3))) i2v*)(l), 0, 0);
#else
    *reinterpret_cast<float2*>(l) = *reinterpret_cast<const float2*>(g);
#endif
}

// copy 4 bytes global -> LDS
__device__ __forceinline__ void lds_copy_b32(const float* g, float* l)
{
#if defined(ASYNC_B32)
    __builtin_amdgcn_global_load_async_to_lds_b32(
        (__attribute__((address_space(1))) int*)(g),
        (__attribute__((address_space(3))) int*)(l), 0, 0);
#else
    *l = *g;
#endif
}

// drain this wave's outstanding async copies (ASYNCcnt -> 0)
__device__ __forceinline__ void lds_copy_drain()
{
#if defined(ASYNC_B64) || defined(ASYNC_B32)
    __builtin_amdgcn_s_wait_asynccnt(0);
#endif
}

// =====================================================================
// Kernel 1: fused im2col GEMM (conv_out + qkv) -> attention -> 1x1 conv
//           writes ReLU'd flattened activations y[B][1600]
// =====================================================================
__global__ __launch_bounds__(256)
void conv_attn_kernel(const float* __restrict__ x,
                      const float* __restrict__ conv_w, const float* __restrict__ conv_b,
                      const float* __restrict__ qkv_w,  const float* __restrict__ qkv_b,
                      const float* __restrict__ attn_w, const float* __restrict__ attn_b,
                      float* __restrict__ y)
{
    __shared__ union {
        float patch[NCOLP * PST];              // phase 1: im2col patches [col][k]
        struct {
            float qkv [GIMG][96][HW];          // phase 2: q(0..31) k(32..63) v(64..95)
            float attnO[GIMG][32][HW];         // attention output per head
        } p2;
    } sh;

    const int tid   = threadIdx.x;
    const int lane  = tid & 31;
    const int wv    = tid >> 5;                // wave id 0..7  -> M tile (16 channels)
    const int bBase = blockIdx.x * GIMG;

    // ---- phase 1a: build im2col patches in LDS  (P[col][k] = x[b,cin,oy+ky,ox+kx])
    //      real columns stream in via async global->LDS DMA; pad columns zero-fill
    for (int idx = tid; idx < NCOLP * KCONV; idx += 256) {
        const int col = idx >> 6;              // /64
        const int k   = idx & 63;
        float* dst = &sh.patch[col * PST + k];
        if (col < COLS) {
            const int img = col / HW, p = col % HW;
            const int oy = p / HOUT, ox = p % HOUT;
            const int cin = k >> 2, ky = (k >> 1) & 1, kx = k & 1;
            lds_copy_b32(
                &x[(((size_t)(bBase + img) * CIN + cin) * HIN + (oy + ky)) * HIN + (ox + kx)],
                dst);
        } else {
            *dst = 0.f;
        }
    }
    lds_copy_drain();
    __syncthreads();

    // ---- phase 1b: load weight A-fragments straight from global
    // combined weight row m: m<32 -> conv_w[m][64], else qkv_w[m-32][64]
    // A layout: lane(M)=lane&15, VGPR pair = (K, K+1), lane-half selects K+2
    const int mrow = wv * 16 + (lane & 15);
    const float* wrowp = (mrow < 32) ? (conv_w + (size_t)mrow * KCONV)
                                     : (qkv_w + (size_t)(mrow - 32) * KCONV);
    v2f aF[16];
#pragma unroll
    for (int s = 0; s < 16; ++s) {
        const int k0 = s * 4 + ((lane >> 4) << 1);
        const float2 t = *reinterpret_cast<const float2*>(wrowp + k0);
        aF[s].x = t.x; aF[s].y = t.y;
    }

    // ---- phase 1c: 7 N-tiles x 16 K-steps of V_WMMA_F32_16X16X4_F32
    v8f acc[7];
#pragma unroll
    for (int nt = 0; nt < 7; ++nt) {
        v8f c = {};
        const int col = nt * 16 + (lane & 15);
#pragma unroll
        for (int s = 0; s < 16; ++s) {
            const int k0 = s * 4 + ((lane >> 4) << 1);
            const float2 t = *reinterpret_cast<const float2*>(&sh.patch[col * PST + k0]);
            v2f bv; bv.x = t.x; bv.y = t.y;
            c = __builtin_amdgcn_wmma_f32_16x16x4_f32(false, aF[s], false, bv,
                                                      (short)0, c, false, false);
        }
        acc[nt] = c;
    }
    __syncthreads();   // all waves done reading sh.patch

    // ---- phase 1d: writeout. conv channels -> global y (ReLU), qkv -> LDS
    {
#pragma unroll
        for (int nt = 0; nt < 7; ++nt) {
            const int col = nt * 16 + (lane & 15);
            if (col >= COLS) continue;
            const int img = col / HW, p = col % HW;
#pragma unroll
            for (int r = 0; r < 8; ++r) {
                const int ch = wv * 16 + ((lane >> 4) ? 8 : 0) + r;   // C/D layout
                float v = acc[nt][r];
                if (ch < 32) {
                    v += conv_b[ch];
                    y[(size_t)(bBase + img) * 1600 + ch * HW + p] = fmaxf(v, 0.f);
                } else {
                    v += qkv_b[ch - 32];                              // dkh=1 => q scale 1.0
                    sh.p2.qkv[img][ch - 32][p] = v;
                }
            }
        }
    }
    __syncthreads();

    // ---- phase 2: attention.  128 (img,head) tasks, 2 threads each (split i)
    {
        const int task = tid >> 1;             // 0..127
        const int half = tid & 1;
        const int img = task >> 5, hd = task & 31;
        float kk[HW], vv[HW];
        float kmax = -3.0e38f, kmin = 3.0e38f;
#pragma unroll
        for (int j = 0; j < HW; ++j) {
            kk[j] = sh.p2.qkv[img][32 + hd][j];
            vv[j] = sh.p2.qkv[img][64 + hd][j];
            kmax = fmaxf(kmax, kk[j]);
            kmin = fminf(kmin, kk[j]);
        }
        for (int i = half; i < HW; i += 2) {
            const float qi = sh.p2.qkv[img][hd][i];
            const float m  = (qi >= 0.f) ? qi * kmax : qi * kmin;  // = max_j qi*k_j
            float den = 0.f, num = 0.f;
#pragma unroll
            for (int j = 0; j < HW; ++j) {
                const float e = __expf(qi * kk[j] - m);
                den += e;
                num += e * vv[j];
            }
            sh.p2.attnO[img][hd][i] = num / den;
        }
    }
    __syncthreads();

    // ---- phase 3: 1x1 attention conv over heads + ReLU -> y channels 32..63
    for (int idx = tid; idx < GIMG * 32 * HW; idx += 256) {
        const int img = idx / (32 * HW);
        const int rem = idx % (32 * HW);
        const int c = rem / HW, p = rem % HW;
        float a = attn_b[c];
#pragma unroll
        for (int h = 0; h < 32; ++h)
            a += attn_w[c * 32 + h] * sh.p2.attnO[img][h][p];
        y[(size_t)(bBase + img) * 1600 + (32 + c) * HW + p] = fmaxf(a, 0.f);
    }
}

// =====================================================================
// Kernel 2: generic fp32 WMMA GEMM  Out[M][N] = act(In[M][K] @ Wt[N][K]^T + b)
//           8 waves/WG, one 16x16 tile per wave, K staged in 64-wide blocks
//           via async global->LDS DMA (ASYNCcnt) when available
// =====================================================================
template <int K, int N, bool RELU>
__global__ __launch_bounds__(256)
void dense_wmma(const float* __restrict__ In, const float* __restrict__ Wt,
                const float* __restrict__ bias, float* __restrict__ Out)
{
    constexpr int NT   = N / 16;       // n-tiles per row-block
    constexpr int MT   = 8 / NT;       // m-tiles per workgroup
    constexpr int ROWS = MT * 16;
    constexpr int KB   = 64;

    __shared__ float sIn[ROWS * PST];
    __shared__ float sWt[N * PST];

    const int tid   = threadIdx.x;
    const int lane  = tid & 31;
    const int wv    = tid >> 5;
    const int ntile = wv % NT;
    const int mtile = wv / NT;
    const int mBase = blockIdx.x * ROWS;

    v8f acc = {};
    for (int kb = 0; kb < K; kb += KB) {
        // stage input rows (float2 granularity, 8B aligned via PST=66)
        for (int i = tid; i < ROWS * (KB / 2); i += 256) {
            const int r = i / (KB / 2), q = i % (KB / 2);
            lds_copy_b64(In + (size_t)(mBase + r) * K + kb + q * 2,
                         &sIn[r * PST + q * 2]);
        }
        // stage weight rows
        for (int i = tid; i < N * (KB / 2); i += 256) {
            const int r = i / (KB / 2), q = i % (KB / 2);
            lds_copy_b64(Wt + (size_t)r * K + kb + q * 2,
                         &sWt[r * PST + q * 2]);
        }
        // prefetch next K-block of the input tile into cache
        if (kb + KB < K && tid < ROWS)
            __builtin_prefetch(In + (size_t)(mBase + tid) * K + kb + KB, 0, 0);
        lds_copy_drain();
        __syncthreads();

#pragma unroll
        for (int s = 0; s < 16; ++s) {
            const int k0 = s * 4 + ((lane >> 4) << 1);
            const float2 ta = *reinterpret_cast<const float2*>(
                &sIn[(mtile * 16 + (lane & 15)) * PST + k0]);
            const float2 tb = *reinterpret_cast<const float2*>(
                &sWt[(ntile * 16 + (lane & 15)) * PST + k0]);
            v2f av, bv;
            av.x = ta.x; av.y = ta.y;
            bv.x = tb.x; bv.y = tb.y;
            acc = __builtin_amdgcn_wmma_f32_16x16x4_f32(false, av, false, bv,
                                                        (short)0, acc, false, false);
        }
        __syncthreads();   // protect LDS reuse (WAR) for next k-block
    }

    // epilogue: bias + optional ReLU, C/D layout
    const int col  = ntile * 16 + (lane & 15);
    const float bc = bias[col];
#pragma unroll
    for (int r = 0; r < 8; ++r) {
        const int row = mBase + mtile * 16 + ((lane >> 4) ? 8 : 0) + r;
        float v = acc[r] + bc;
        if (RELU) v = fmaxf(v, 0.f);
        Out[(size_t)row * N + col] = v;
    }
}

// =====================================================================
// Kernel 3: tiny final dense  out[B][5] = h2[B][64] @ w3[5][64]^T + b3
// =====================================================================
__global__ __launch_bounds__(256)
void dense_out_kernel(const float* __restrict__ In, const float* __restrict__ W,
                      const float* __restrict__ bias, float* __restrict__ Out)
{
    const int idx = blockIdx.x * blockDim.x + threadIdx.x;
    if (idx >= BATCH * OUTCH) return;
    const int b = idx / OUTCH, c = idx % OUTCH;
    const float* r = In + (size_t)b * 64;
    const float* w = W + c * 64;
    float a = bias[c];
#pragma unroll
    for (int k = 0; k < 64; ++k) a += r[k] * w[k];
    Out[idx] = a;
}

// =====================================================================
extern "C" void kernel_launch(void* const* d_in, const int* in_sizes, int n_in,
                              void* d_out, int out_size, void* d_ws, size_t ws_size,
                              hipStream_t stream)
{
    const float* x      = (const float*)d_in[0];
    const float* conv_w = (const float*)d_in[1];
    const float* conv_b = (const float*)d_in[2];
    const float* qkv_w  = (const float*)d_in[3];
    const float* qkv_b  = (const float*)d_in[4];
    const float* attn_w = (const float*)d_in[5];
    const float* attn_b = (const float*)d_in[6];
    const float* w1     = (const float*)d_in[7];
    const float* b1     = (const float*)d_in[8];
    const float* w2     = (const float*)d_in[9];
    const float* b2     = (const float*)d_in[10];
    const float* w3     = (const float*)d_in[11];
    const float* b3     = (const float*)d_in[12];
    float* out = (float*)d_out;

    // workspace layout
    float* y  = (float*)d_ws;                       // [B][1600]
    float* h1 = y  + (size_t)BATCH * 1600;          // [B][128]
    float* h2 = h1 + (size_t)BATCH * HID;           // [B][64]

    // 1) fused conv + attention -> flattened ReLU activations
    conv_attn_kernel<<<BATCH / GIMG, 256, 0, stream>>>(
        x, conv_w, conv_b, qkv_w, qkv_b, attn_w, attn_b, y);

    // 2) dense1: [B,1600] x [1600,128]^T, ReLU
    dense_wmma<1600, 128, true><<<BATCH / 16, 256, 0, stream>>>(y, w1, b1, h1);

    // 3) dense2: [B,128] x [128,64]^T, ReLU
    dense_wmma<128, 64, true><<<BATCH / 32, 256, 0, stream>>>(h1, w2, b2, h2);

    // 4) dense3: [B,64] x [64,5]^T
    dense_out_kernel<<<(BATCH * OUTCH + 255) / 256, 256, 0, stream>>>(h2, w3, b3, out);
}